// get_LL_mnist_torch_5368709120808
// MI455X (gfx1250) — compile-verified
//
#include <hip/hip_runtime.h>
#include <cstdint>

// Closed form of 2-level Haar DWT (db1, zero mode) with high bands zeroed,
// then 2-level inverse:  out[4x4 block] = (1/16) * sum(in[4x4 block]),
// broadcast over the block. Pure streaming kernel: ~205 MB traffic,
// ~8.8 us floor at 23.3 TB/s HBM. Uses CDNA5 async global<->LDS copies so
// both HBM directions are perfectly linear b128 streams; the 112B-strided
// block access pattern is served out of LDS.

#define THREADS        256
#define IMGS_PER_WG    16
#define FLOATS_PER_IMG 784                              // 28*28
#define CHUNKS_PER_IMG 49                               // 4x4 blocks / image (784/16)
#define TILE_FLOATS    (IMGS_PER_WG * FLOATS_PER_IMG)   // 12544 floats = 50176 B LDS

__device__ __forceinline__ uint32_t lds_addr32(const void* p) {
  // addrspace(3) -> flat keeps the LDS byte offset in the low 32 bits.
  return (uint32_t)(uintptr_t)p;
}

__global__ __launch_bounds__(THREADS)
void haar_ll2_fwd_inv_kernel(const float* __restrict__ in,
                             float* __restrict__ out,
                             int numImages) {
  __shared__ float tile[TILE_FLOATS];

  const int tid      = (int)threadIdx.x;
  const int imgBase  = (int)blockIdx.x * IMGS_PER_WG;
  int imgsHere = numImages - imgBase;
  if (imgsHere <= 0) return;
  if (imgsHere > IMGS_PER_WG) imgsHere = IMGS_PER_WG;

  const int    chunks     = imgsHere * CHUNKS_PER_IMG;      // also == #blocks
  const size_t gFloatBase = (size_t)imgBase * FLOATS_PER_IMG;
  const uint32_t ldsBase  = lds_addr32(&tile[0]);

  // ---- Stage tile: HBM -> LDS, linear async b128 copies (ASYNCcnt path) ----
  for (int c = tid; c < chunks; c += THREADS) {
    uint32_t           l = ldsBase + (uint32_t)c * 16u;
    unsigned long long g =
        (unsigned long long)(const void*)(in + gFloatBase + (size_t)c * 4);
    asm volatile("global_load_async_to_lds_b128 %0, %1, off"
                 :: "v"(l), "v"(g) : "memory");
  }
  asm volatile("s_wait_asynccnt 0" ::: "memory");
  __syncthreads();

  // ---- Compute in place: one thread per 4x4 block (49 blocks/image) ----
  for (int n = tid; n < chunks; n += THREADS) {
    int img = n / 49;
    int blk = n - img * 49;
    int by  = blk / 7;
    int bx  = blk - by * 7;
    float* lp = &tile[img * FLOATS_PER_IMG + by * (4 * 28) + bx * 4];

    float4 r0 = *(const float4*)(lp + 0 * 28);
    float4 r1 = *(const float4*)(lp + 1 * 28);
    float4 r2 = *(const float4*)(lp + 2 * 28);
    float4 r3 = *(const float4*)(lp + 3 * 28);

    float s = (((r0.x + r0.y) + (r0.z + r0.w)) +
               ((r1.x + r1.y) + (r1.z + r1.w))) +
              (((r2.x + r2.y) + (r2.z + r2.w)) +
               ((r3.x + r3.y) + (r3.z + r3.w)));
    s *= 0.0625f;

    float4 v = make_float4(s, s, s, s);
    *(float4*)(lp + 0 * 28) = v;   // block's output region == its input region,
    *(float4*)(lp + 1 * 28) = v;   // so in-place is race-free per thread
    *(float4*)(lp + 2 * 28) = v;
    *(float4*)(lp + 3 * 28) = v;
  }
  __syncthreads();

  // ---- Drain tile: LDS -> HBM, linear async b128 stores ----
  for (int c = tid; c < chunks; c += THREADS) {
    uint32_t           l = ldsBase + (uint32_t)c * 16u;
    unsigned long long g =
        (unsigned long long)(void*)(out + gFloatBase + (size_t)c * 4);
    asm volatile("global_store_async_from_lds_b128 %0, %1, off"
                 :: "v"(g), "v"(l) : "memory");
  }
  // S_ENDPGM performs an implicit wait-idle, draining ASYNCcnt.
}

extern "C" void kernel_launch(void* const* d_in, const int* in_sizes, int n_in,
                              void* d_out, int out_size, void* d_ws, size_t ws_size,
                              hipStream_t stream) {
  // d_in[0]: float32 image [32768,1,28,28]; d_in[1]: lev (==2, static in setup).
  const float* in  = (const float*)d_in[0];
  float*       out = (float*)d_out;

  const int numImages = in_sizes[0] / FLOATS_PER_IMG;   // 32768
  const int grid      = (numImages + IMGS_PER_WG - 1) / IMGS_PER_WG;   // 2048

  haar_ll2_fwd_inv_kernel<<<grid, THREADS, 0, stream>>>(in, out, numImages);
}